// GridMaxPoolReduction_13408887898338
// MI455X (gfx1250) — compile-verified
//
#include <hip/hip_runtime.h>
#include <cstdint>

// ---------------------------------------------------------------------------
// GridMaxPool: segment-max over (event, 16x16 cell) + gather back to points.
// Bandwidth-bound (~420 MB @ 23.3 TB/s ~= 18 us). No matmul -> no WMMA; use
// CDNA5 ds_max_num_f32 / global_atomic_max_num_f32 / async LDS->global store.
// ---------------------------------------------------------------------------

#define GRID_DIM 16
#define CELLS    256      // 16*16
#define F        128      // features per point
#define FQ       32       // features per quarter (one lane each, wave32)
#define NQ       4        // feature quarters
#define CP       4096     // points per scatter chunk (<= any event length)

__device__ __forceinline__ unsigned lds_off(const void* p) {
  // Generic pointer to LDS: low 32 bits are the wave-relative LDS byte offset
  // (ISA 10.2 aperture table).
  return (unsigned)(unsigned long long)(uintptr_t)p;
}

__device__ __forceinline__ void lds_max_f32(unsigned off, float v) {
  asm volatile("ds_max_num_f32 %0, %1" :: "v"(off), "v"(v) : "memory");
}

__device__ __forceinline__ void global_max_f32(float* p, float v) {
  asm volatile("global_atomic_max_num_f32 %0, %1, off"
               :: "v"((unsigned long long)(uintptr_t)p), "v"(v) : "memory");
}

__device__ __forceinline__ int cell_of(float c) {
  int x = (int)floorf(c) + 8;
  x = x < 0 ? 0 : x;
  x = x > GRID_DIM - 1 ? GRID_DIM - 1 : x;
  return x;
}

// ---------------------------------------------------------------- init table
__global__ void __launch_bounds__(256)
gmp_init_kernel(float4* __restrict__ tab4, int total4) {
  int i = blockIdx.x * 256 + threadIdx.x;
  float ninf = -__builtin_inff();
  if (i < total4) tab4[i] = make_float4(ninf, ninf, ninf, ninf);
}

// ------------------------------------------------------------- scatter (max)
// grid = (ceil(N/CP), NQ), block = 256 (8 waves). One wave handles one point
// per iteration; lane k owns feature (q*32 + k). LDS table: 2 local events x
// 256 cells x 32 feats = 64 KB; stride-4B lane addresses -> bank-conflict-free.
__global__ void __launch_bounds__(256)
gmp_scatter_kernel(const float* __restrict__ coords,
                   const float* __restrict__ feat,
                   const int*   __restrict__ rs,
                   float*       __restrict__ tabg,
                   int N, int n_events) {
  __shared__ float tab[2 * CELLS * FQ];   // 16384 floats = 64 KB

  const int tid  = threadIdx.x;
  const int lane = tid & 31;
  const int wave = tid >> 5;
  const int q    = blockIdx.y;
  const int fb   = q * FQ;
  const int base = blockIdx.x * CP;
  const int end  = (base + CP < N) ? (base + CP) : N;

  // event of first point in chunk
  int e0 = 0;
  for (int j = 1; j < n_events; ++j)
    if (base >= rs[j]) e0 = j;

  // cache up to 8 event boundaries in registers (rs is tiny & L2-hot)
  int rb[8];
#pragma unroll
  for (int j = 0; j < 8; ++j)
    rb[j] = (j + 1 <= n_events) ? rs[j + 1] : 0x7fffffff;

  for (int i = tid; i < 2 * CELLS * FQ; i += 256) tab[i] = -__builtin_inff();
  __syncthreads();

  for (int p = base + wave; p < end; p += 8) {
    const float c0 = coords[2 * p];
    const float c1 = coords[2 * p + 1];
    const int cell = cell_of(c0) * GRID_DIM + cell_of(c1);

    int ev = 0;
#pragma unroll
    for (int j = 0; j < 8; ++j)
      if (p >= rb[j]) ev = j + 1;
    for (int j = 9; j < n_events; ++j)   // >8 events: rare tail path
      if (p >= rs[j]) ev = j;

    const float v = feat[(size_t)p * F + fb + lane];
    if (p + 8 < end)
      __builtin_prefetch(&feat[(size_t)(p + 8) * F + fb + lane], 0, 0);

    const int le = ev - e0;              // 0 or 1 when chunk <= event length
    if (le < 2) {
      lds_max_f32(lds_off(&tab[(le * CELLS + cell) * FQ + lane]), v);
    } else {
      // correctness fallback for arbitrary row splits
      global_max_f32(&tabg[((size_t)ev * CELLS + cell) * F + fb + lane], v);
    }
  }
  __syncthreads();

  // flush non-empty entries; stride-1 over k -> conflict-free LDS reads
  for (int i = tid; i < 2 * CELLS * FQ; i += 256) {
    const float v = tab[i];
    if (v > -3.0e38f) {
      const int k     = i & (FQ - 1);
      const int lcell = i >> 5;                 // /FQ
      const int ev    = e0 + (lcell >> 8);
      const int cell  = lcell & (CELLS - 1);
      if (ev < n_events)
        global_max_f32(&tabg[((size_t)ev * CELLS + cell) * F + fb + k], v);
    }
  }
}

// ------------------------------------------------------- gather + fixup + out
// One point per wave; lane holds 4 feats (b128). Fixed-up row is staged in LDS
// (double-buffered) and streamed out with CDNA5 async LDS->global b128 stores.
__global__ void __launch_bounds__(256)
gmp_gather_kernel(const float* __restrict__ coords,
                  const int*   __restrict__ rs,
                  const float* __restrict__ tabg,
                  float*       __restrict__ out,
                  int N, int n_events) {
  __shared__ float4 stage[2][8][32];   // [buf][wave][lane] = 8 KB

  const int tid  = threadIdx.x;
  const int lane = tid & 31;
  const int wave = tid >> 5;

  int rb[8];
#pragma unroll
  for (int j = 0; j < 8; ++j)
    rb[j] = (j + 1 <= n_events) ? rs[j + 1] : 0x7fffffff;

  const int wid0   = blockIdx.x * 8 + wave;
  const int nwaves = gridDim.x * 8;

  int it = 0;
  for (int p = wid0; p < N; p += nwaves, ++it) {
    const float c0 = coords[2 * p];
    const float c1 = coords[2 * p + 1];
    const int cell = cell_of(c0) * GRID_DIM + cell_of(c1);

    int ev = 0;
#pragma unroll
    for (int j = 0; j < 8; ++j)
      if (p >= rb[j]) ev = j + 1;
    for (int j = 9; j < n_events; ++j)
      if (p >= rs[j]) ev = j;

    const size_t srow = ((size_t)ev * CELLS + cell) * F;   // L2-resident row
    float4 v = *(const float4*)&tabg[srow + lane * 4];
    v.x = (fabsf(v.x) > 1.0e10f) ? 0.0f : v.x;
    v.y = (fabsf(v.y) > 1.0e10f) ? 0.0f : v.y;
    v.z = (fabsf(v.z) > 1.0e10f) ? 0.0f : v.z;
    v.w = (fabsf(v.w) > 1.0e10f) ? 0.0f : v.w;

    const int buf = it & 1;
    // async store issued 2 iterations ago (same buffer) must be done
    asm volatile("s_wait_asynccnt 0x1" ::: "memory");
    stage[buf][wave][lane] = v;                        // ds_store_b128
    asm volatile("s_wait_dscnt 0x0" ::: "memory");     // data visible in LDS

    const unsigned long long ga =
        (unsigned long long)(uintptr_t)&out[(size_t)p * F + lane * 4];
    const unsigned la = lds_off(&stage[buf][wave][lane]);
    asm volatile("global_store_async_from_lds_b128 %0, %1, off"
                 :: "v"(ga), "v"(la) : "memory");
  }
  asm volatile("s_wait_asynccnt 0x0" ::: "memory");
}

// ---------------------------------------------------------------------------
extern "C" void kernel_launch(void* const* d_in, const int* in_sizes, int n_in,
                              void* d_out, int out_size, void* d_ws, size_t ws_size,
                              hipStream_t stream) {
  const float* coords = (const float*)d_in[0];
  const float* feat   = (const float*)d_in[1];
  const int*   rs     = (const int*)d_in[2];
  float*       out    = (float*)d_out;

  const int N        = in_sizes[0] / 2;
  const int n_events = in_sizes[2] - 1;

  float* tabg = (float*)d_ws;                       // [n_events*256][128] f32
  const int total  = n_events * CELLS * F;          // 131072 for 4 events
  const int total4 = total / 4;

  gmp_init_kernel<<<(total4 + 255) / 256, 256, 0, stream>>>(
      (float4*)tabg, total4);

  dim3 sgrid((N + CP - 1) / CP, NQ);
  gmp_scatter_kernel<<<sgrid, 256, 0, stream>>>(coords, feat, rs, tabg, N,
                                                n_events);

  int gblocks = (N + 7) / 8;
  if (gblocks > 4096) gblocks = 4096;               // ~13 pts/wave -> pipelined
  gmp_gather_kernel<<<gblocks, 256, 0, stream>>>(coords, rs, tabg, out, N,
                                                 n_events);
}